// RadialSpectrumCalculator_54700703482416
// MI455X (gfx1250) — compile-verified
//
#include <hip/hip_runtime.h>

// RadialSpectrumCalculator for gfx1250 (MI455X).
//
// Per edge e: r=|v|, x=min(r,5)/5, th=pi*x, s_k=sin(k*th) (k=1..8 via
// Chebyshev recursion from sin/cos), g=0.5*(cos th + 1)/(r+1e-7).
// Scatter-add s_{n+l}*g into out[center*104 + off_l + sp*N_MAX[l] + (n-1)].
//
// Output layout (f32, [n_centers, 104]):
//   l=0: cols  0..31  = sp*8 + (n-1), n=1..8
//   l=1: cols 32..59  = 32+sp*7+(n-1), n=1..7
//   l=2: cols 60..83  = 60+sp*6+(n-1), n=1..6
//   l=3: cols 84..103 = 84+sp*5+(n-1), n=1..5
//
// Bottleneck: 52M f32 atomics into a 20.8MB L2-resident accumulator ->
// use no-return agent-scope fadd (STOREcnt, fire-and-forget) + prefetch.

#define ROW_F 104

__device__ __forceinline__ void atom_add_nr(float* p, float v) {
    // Relaxed agent-scope fadd; result unused -> no-return global_atomic_add_f32.
    (void)__hip_atomic_fetch_add(p, v, __ATOMIC_RELAXED, __HIP_MEMORY_SCOPE_AGENT);
}

__global__ __launch_bounds__(256)
void radial_scatter_kernel(const float* __restrict__ ev,     // [E,3]
                           const int*   __restrict__ cidx,   // [E]
                           const int*   __restrict__ spc,    // [E]
                           float*       __restrict__ out,    // [C,104]
                           int E) {
    const float PI     = 3.14159265358979323846f;
    const float INV_RC = 0.2f;      // 1/5.0
    const float RC     = 5.0f;
    const float EPS    = 1e-7f;

    int i      = blockIdx.x * blockDim.x + threadIdx.x;
    int stride = gridDim.x * blockDim.x;

    for (; i < E; i += stride) {
        // gfx1250 software prefetch of next iteration's stream data.
        int inext = i + stride;
        if (inext < E) {
            __builtin_prefetch(&ev[3 * (size_t)inext], 0, 0);
            __builtin_prefetch(&cidx[inext], 0, 0);
            __builtin_prefetch(&spc[inext], 0, 0);
        }

        // Issue all loads for this edge up front.
        size_t b3 = 3 * (size_t)i;
        float vx = ev[b3 + 0];
        float vy = ev[b3 + 1];
        float vz = ev[b3 + 2];
        int   c  = cidx[i];
        int   sp = spc[i];

        float r2 = fmaf(vx, vx, fmaf(vy, vy, vz * vz));
        float r  = sqrtf(r2);
        float x  = fminf(r, RC) * INV_RC;        // in [0,1]
        float th = PI * x;

        float s1 = __sinf(th);
        float c1 = __cosf(th);

        // g = fcut/(r+eps) = 0.5*(cos th + 1)/(r+eps)
        float g  = (0.5f * (c1 + 1.0f)) * __frcp_rn(r + EPS);

        // Chebyshev: s_{k+1} = 2c*s_k - s_{k-1}; 8 sines from 2 transcendentals.
        float t  = c1 + c1;
        float s2 = t * s1;
        float s3 = fmaf(t, s2, -s1);
        float s4 = fmaf(t, s3, -s2);
        float s5 = fmaf(t, s4, -s3);
        float s6 = fmaf(t, s5, -s4);
        float s7 = fmaf(t, s6, -s5);
        float s8 = fmaf(t, s7, -s6);

        float v1 = s1 * g, v2 = s2 * g, v3 = s3 * g, v4 = s4 * g;
        float v5 = s5 * g, v6 = s6 * g, v7 = s7 * g, v8 = s8 * g;

        float* row = out + (size_t)c * ROW_F;

        // l = 0 : values s_1..s_8
        float* p0 = row + (sp << 3);
        atom_add_nr(p0 + 0, v1); atom_add_nr(p0 + 1, v2);
        atom_add_nr(p0 + 2, v3); atom_add_nr(p0 + 3, v4);
        atom_add_nr(p0 + 4, v5); atom_add_nr(p0 + 5, v6);
        atom_add_nr(p0 + 6, v7); atom_add_nr(p0 + 7, v8);

        // l = 1 : values s_2..s_8
        float* p1 = row + 32 + sp * 7;
        atom_add_nr(p1 + 0, v2); atom_add_nr(p1 + 1, v3);
        atom_add_nr(p1 + 2, v4); atom_add_nr(p1 + 3, v5);
        atom_add_nr(p1 + 4, v6); atom_add_nr(p1 + 5, v7);
        atom_add_nr(p1 + 6, v8);

        // l = 2 : values s_3..s_8
        float* p2 = row + 60 + sp * 6;
        atom_add_nr(p2 + 0, v3); atom_add_nr(p2 + 1, v4);
        atom_add_nr(p2 + 2, v5); atom_add_nr(p2 + 3, v6);
        atom_add_nr(p2 + 4, v7); atom_add_nr(p2 + 5, v8);

        // l = 3 : values s_4..s_8
        float* p3 = row + 84 + sp * 5;
        atom_add_nr(p3 + 0, v4); atom_add_nr(p3 + 1, v5);
        atom_add_nr(p3 + 2, v6); atom_add_nr(p3 + 3, v7);
        atom_add_nr(p3 + 4, v8);
    }
}

// Branch-minimal memset: one element per thread, no grid-stride division mess.
__global__ __launch_bounds__(256)
void zero_out_kernel(float* __restrict__ out, int n) {
    int i = blockIdx.x * blockDim.x + threadIdx.x;
    if (i < n) out[i] = 0.0f;
}

extern "C" void kernel_launch(void* const* d_in, const int* in_sizes, int n_in,
                              void* d_out, int out_size, void* d_ws, size_t ws_size,
                              hipStream_t stream) {
    (void)n_in; (void)d_ws; (void)ws_size;

    const float* ev = (const float*)d_in[0];   // edge_vec  [E,3] f32
    const int*   ci = (const int*)d_in[1];     // center_index [E] i32
    const int*   sp = (const int*)d_in[2];     // neighbor_species [E] i32
    // d_in[3] = n_centers (scalar); out_size == n_centers*104 already.

    int E = in_sizes[1];
    float* out = (float*)d_out;

    // 1) Zero the accumulator (d_out is poisoned; we accumulate atomically).
    {
        int threads = 256;
        int blocks  = (out_size + threads - 1) / threads;
        if (blocks < 1) blocks = 1;
        zero_out_kernel<<<blocks, threads, 0, stream>>>(out, out_size);
    }

    // 2) Atomic scatter. Grid-stride persistent blocks (~4 edges/thread) so
    //    the software prefetch has a next iteration to target.
    {
        int threads = 256;
        long long want = ((long long)E + threads * 4 - 1) / (threads * 4);
        int blocks = (int)(want < 1 ? 1 : (want > 16384 ? 16384 : want));
        radial_scatter_kernel<<<blocks, threads, 0, stream>>>(ev, ci, sp, out, E);
    }
}